// MiniMaxM2SparseMoeBlock_7945689498098
// MI455X (gfx1250) — compile-verified
//
#include <hip/hip_runtime.h>
#include <math.h>

// MiniMax-M2 sparse MoE: T=1024, H=2048, I=1024, E=16, top-2. fp32 throughout.
// HBM-bound: 384 MB fp32 expert weights / 23.3 TB/s ~= 16.5 us floor.
// M_TILE=64 amortizes the B-fragment stream over 4 WMMA row-tiles and cuts
// per-expert weight re-reads to ~2x (absorbed by 192 MB L2).

#define T_TOK 1024
#define H_DIM 2048
#define I_DIM 1024
#define E_NUM 16

#define M_TILE   64                 // tokens per block (4 x 16-row WMMA tiles)
#define MT_NUM   (T_TOK / M_TILE)   // 16 m-tiles cover worst case
#define NSPLIT1  8                  // GEMM1: 64 I-tiles -> 8 blocks x 8 waves x 1 tile
#define NSPLIT2  16                 // GEMM2: 128 H-tiles -> 16 blocks x 8 waves x 1 tile

typedef __attribute__((ext_vector_type(2))) float v2f;
typedef __attribute__((ext_vector_type(8))) float v8f;

// ---------------------------------------------------------------------------
// Kernel 1: router. One wave32 per token. logits -> softmax -> top2(scores+bias)
// -> renormalized raw-score weights -> scatter (token|slot<<16, weight) into
// per-expert buckets. Bucket order is atomic-nondeterministic, but per-token
// math is position-independent -> bitwise-deterministic final output.
// ---------------------------------------------------------------------------
__global__ void moe_router(const float* __restrict__ x,
                           const float* __restrict__ gw,
                           const float* __restrict__ bias,
                           int* __restrict__ counts,
                           int* __restrict__ bucket_tok,
                           float* __restrict__ bucket_w) {
    const int lane = threadIdx.x & 31;
    const int wid  = threadIdx.x >> 5;
    const int t = blockIdx.x * 8 + wid;
    if (t >= T_TOK) return;

    const float* xr = x + (size_t)t * H_DIM;
    float logit[E_NUM];
    #pragma unroll
    for (int e = 0; e < E_NUM; ++e) {
        const float* gr = gw + (size_t)e * H_DIM;
        float s = 0.f;
        for (int h = lane; h < H_DIM; h += 32)
            s = fmaf(xr[h], gr[h], s);
        #pragma unroll
        for (int off = 16; off >= 1; off >>= 1)
            s += __shfl_xor(s, off, 32);
        logit[e] = s;   // all lanes hold the full sum
    }

    float mx = logit[0];
    #pragma unroll
    for (int e = 1; e < E_NUM; ++e) mx = fmaxf(mx, logit[e]);
    float sum = 0.f, sc[E_NUM];
    #pragma unroll
    for (int e = 0; e < E_NUM; ++e) { sc[e] = __expf(logit[e] - mx); sum += sc[e]; }
    const float inv = 1.f / sum;
    #pragma unroll
    for (int e = 0; e < E_NUM; ++e) sc[e] *= inv;

    // top-2 on (scores + bias); strict '>' keeps lowest index on ties (lax.top_k)
    int i1 = 0; float b1v = sc[0] + bias[0];
    #pragma unroll
    for (int e = 1; e < E_NUM; ++e) {
        float v = sc[e] + bias[e];
        if (v > b1v) { b1v = v; i1 = e; }
    }
    int i2 = (i1 == 0) ? 1 : 0; float b2v = sc[i2] + bias[i2];
    #pragma unroll
    for (int e = 0; e < E_NUM; ++e) {
        if (e == i1) continue;
        float v = sc[e] + bias[e];
        if (v > b2v) { b2v = v; i2 = e; }
    }

    float w1 = sc[i1], w2 = sc[i2];
    const float wn = 1.f / (w1 + w2);
    w1 *= wn; w2 *= wn;

    if (lane == 0) {
        int p1 = atomicAdd(&counts[i1], 1);
        bucket_tok[i1 * T_TOK + p1] = t;                 // slot 0
        bucket_w  [i1 * T_TOK + p1] = w1;
        int p2 = atomicAdd(&counts[i2], 1);
        bucket_tok[i2 * T_TOK + p2] = t | (1 << 16);     // slot 1
        bucket_w  [i2 * T_TOK + p2] = w2;
    }
}

// ---------------------------------------------------------------------------
// Kernel 2a: GEMM1 + SwiGLU. he[(t*2+slot), :] = silu(x@w1e) * (x@w3e).
// Block = 256 thr (8 waves), one (expert, 64-token m-tile, 8 I-tile chunk).
// Each wave owns one 16-col I-tile; 4 x WMMA f32 16x16x4 row-tiles per k-step.
// Fragment layouts (V_WMMA_F32_16X16X4_F32):
//   A 16x4: lane l -> row m=l%16, K=(l/16)*2+{0,1}
//   B 4x16: lane l -> col n=l%16, K=(l/16)*2+{0,1}
//   C/D:    vgpr r, lane l -> (m = r + (l/16)*8, n = l%16)
// ---------------------------------------------------------------------------
__global__ void moe_gemm1(const float* __restrict__ x,
                          const float* __restrict__ w1,
                          const float* __restrict__ w3,
                          const int*   __restrict__ counts,
                          const int*   __restrict__ bucket_tok,
                          float* __restrict__ he_buf) {
    __shared__ int s_tok[M_TILE];
    __shared__ int s_slot[M_TILE];
    __shared__ int s_valid[M_TILE];

    const int e   = blockIdx.z;
    const int mt  = blockIdx.y;
    const int cnt = counts[e];
    const int base = mt * M_TILE;
    if (base >= cnt) return;                 // block-uniform: EXEC stays full for WMMA

    const int tid = threadIdx.x;
    if (tid < M_TILE) {
        int idx = base + tid;
        int v   = (idx < cnt);
        int ic  = v ? idx : (cnt - 1);       // clamp tail rows to a valid token
        int pk  = bucket_tok[e * T_TOK + ic];
        s_tok[tid]   = pk & 0xFFFF;
        s_slot[tid]  = pk >> 16;
        s_valid[tid] = v;
    }
    __syncthreads();

    const int lane  = tid & 31;
    const int wid   = tid >> 5;
    const int mrow  = lane & 15;
    const int koff  = (lane >> 4) << 1;      // 0 or 2
    const int mbase = (lane >> 4) << 3;      // 0 or 8

    const int n0   = (blockIdx.x * 8 + wid) * 16;   // this wave's I-tile
    const int ncol = n0 + mrow;

    const float* w1e = w1 + (size_t)e * H_DIM * I_DIM;
    const float* w3e = w3 + (size_t)e * H_DIM * I_DIM;

    const float* xp[4];
    #pragma unroll
    for (int ms = 0; ms < 4; ++ms)
        xp[ms] = x + (size_t)s_tok[ms * 16 + mrow] * H_DIM + koff;

    v8f acc1[4] = {}, acc3[4] = {};
    for (int k = 0; k < H_DIM; k += 4) {
        v2f b1, b3;
        b1.x = w1e[(size_t)(k + koff    ) * I_DIM + ncol];
        b1.y = w1e[(size_t)(k + koff + 1) * I_DIM + ncol];
        b3.x = w3e[(size_t)(k + koff    ) * I_DIM + ncol];
        b3.y = w3e[(size_t)(k + koff + 1) * I_DIM + ncol];
        #pragma unroll
        for (int ms = 0; ms < 4; ++ms) {
            v2f a = *(const v2f*)(xp[ms] + k);
            acc1[ms] = __builtin_amdgcn_wmma_f32_16x16x4_f32(false, a, false, b1,
                                                             (short)0, acc1[ms], false, false);
            acc3[ms] = __builtin_amdgcn_wmma_f32_16x16x4_f32(false, a, false, b3,
                                                             (short)0, acc3[ms], false, false);
        }
    }

    #pragma unroll
    for (int ms = 0; ms < 4; ++ms) {
        #pragma unroll
        for (int r = 0; r < 8; ++r) {
            int m = ms * 16 + mbase + r;
            if (s_valid[m]) {
                float z = acc1[ms][r];
                float h = (z / (1.f + __expf(-z))) * acc3[ms][r];   // silu(z)*gate
                he_buf[((size_t)s_tok[m] * 2 + s_slot[m]) * I_DIM + ncol] = h;
            }
        }
    }
}

// ---------------------------------------------------------------------------
// Kernel 2b: GEMM2. partial[(t*2+slot), :] = weight * (he @ w2e).
// Block = 256 thr, one (expert, 64-token m-tile, 8 H-tile chunk).
// he_buf (8 MB) is L2-resident, so its re-reads cost ~no DRAM traffic.
// ---------------------------------------------------------------------------
__global__ void moe_gemm2(const float* __restrict__ he_buf,
                          const float* __restrict__ w2,
                          const int*   __restrict__ counts,
                          const int*   __restrict__ bucket_tok,
                          const float* __restrict__ bucket_w,
                          float* __restrict__ partial) {
    __shared__ int   s_tok[M_TILE];
    __shared__ int   s_slot[M_TILE];
    __shared__ float s_w[M_TILE];
    __shared__ int   s_valid[M_TILE];

    const int e   = blockIdx.z;
    const int mt  = blockIdx.y;
    const int cnt = counts[e];
    const int base = mt * M_TILE;
    if (base >= cnt) return;

    const int tid = threadIdx.x;
    if (tid < M_TILE) {
        int idx = base + tid;
        int v   = (idx < cnt);
        int ic  = v ? idx : (cnt - 1);
        int pk  = bucket_tok[e * T_TOK + ic];
        s_tok[tid]   = pk & 0xFFFF;
        s_slot[tid]  = pk >> 16;
        s_w[tid]     = bucket_w[e * T_TOK + ic];
        s_valid[tid] = v;
    }
    __syncthreads();

    const int lane  = tid & 31;
    const int wid   = tid >> 5;
    const int mrow  = lane & 15;
    const int koff  = (lane >> 4) << 1;
    const int mbase = (lane >> 4) << 3;

    const int n0   = (blockIdx.x * 8 + wid) * 16;   // this wave's H-tile
    const int ncol = n0 + mrow;

    const float* w2e = w2 + (size_t)e * I_DIM * H_DIM;

    const float* hp[4];
    #pragma unroll
    for (int ms = 0; ms < 4; ++ms) {
        int m = ms * 16 + mrow;
        hp[ms] = he_buf + ((size_t)s_tok[m] * 2 + s_slot[m]) * I_DIM + koff;
    }

    v8f acc[4] = {};
    for (int k = 0; k < I_DIM; k += 4) {
        v2f b;
        b.x = w2e[(size_t)(k + koff    ) * H_DIM + ncol];
        b.y = w2e[(size_t)(k + koff + 1) * H_DIM + ncol];
        #pragma unroll
        for (int ms = 0; ms < 4; ++ms) {
            v2f a = *(const v2f*)(hp[ms] + k);
            acc[ms] = __builtin_amdgcn_wmma_f32_16x16x4_f32(false, a, false, b,
                                                            (short)0, acc[ms], false, false);
        }
    }

    #pragma unroll
    for (int ms = 0; ms < 4; ++ms) {
        #pragma unroll
        for (int r = 0; r < 8; ++r) {
            int m = ms * 16 + mbase + r;
            if (s_valid[m]) {
                float val = acc[ms][r] * s_w[m];
                partial[((size_t)s_tok[m] * 2 + s_slot[m]) * H_DIM + ncol] = val;
            }
        }
    }
}

// ---------------------------------------------------------------------------
// Kernel 3: combine the two expert slots per token (every slot written once).
// ---------------------------------------------------------------------------
__global__ void moe_combine(const float* __restrict__ partial,
                            float* __restrict__ out) {
    size_t i = (size_t)blockIdx.x * blockDim.x + threadIdx.x;
    if (i >= (size_t)T_TOK * H_DIM) return;
    size_t t = i / H_DIM, h = i - t * H_DIM;
    out[i] = partial[(t * 2) * H_DIM + h] + partial[(t * 2 + 1) * H_DIM + h];
}

extern "C" void kernel_launch(void* const* d_in, const int* in_sizes, int n_in,
                              void* d_out, int out_size, void* d_ws, size_t ws_size,
                              hipStream_t stream) {
    const float* x    = (const float*)d_in[0];   // [T,H]
    const float* gw   = (const float*)d_in[1];   // [E,H]
    const float* bias = (const float*)d_in[2];   // [E]
    const float* w1   = (const float*)d_in[3];   // [E,H,I]
    const float* w3   = (const float*)d_in[4];   // [E,H,I]
    const float* w2   = (const float*)d_in[5];   // [E,I,H]
    float* out = (float*)d_out;                  // [T,H] fp32

    // workspace layout (all 16B-aligned)
    char*  ws         = (char*)d_ws;
    int*   counts     = (int*)ws;                                     // E ints
    int*   bucket_tok = (int*)(ws + 4096);                            // E*T
    float* bucket_w   = (float*)(ws + 4096 + (size_t)E_NUM * T_TOK * 4);
    float* he_buf     = (float*)(ws + 4096 + (size_t)2 * E_NUM * T_TOK * 4);          // T*2*I
    float* partial    = (float*)(ws + 4096 + (size_t)2 * E_NUM * T_TOK * 4
                                          + (size_t)T_TOK * 2 * I_DIM * 4);           // T*2*H

    hipMemsetAsync(counts, 0, E_NUM * sizeof(int), stream);

    moe_router<<<T_TOK / 8, 256, 0, stream>>>(x, gw, bias, counts, bucket_tok, bucket_w);

    dim3 g1(NSPLIT1, MT_NUM, E_NUM);
    moe_gemm1<<<g1, 256, 0, stream>>>(x, w1, w3, counts, bucket_tok, he_buf);

    dim3 g2(NSPLIT2, MT_NUM, E_NUM);
    moe_gemm2<<<g2, 256, 0, stream>>>(he_buf, w2, counts, bucket_tok, bucket_w, partial);

    moe_combine<<<((size_t)T_TOK * H_DIM + 255) / 256, 256, 0, stream>>>(partial, out);
}